// HermiteActivation_21758304321931
// MI455X (gfx1250) — compile-verified
//
#include <hip/hip_runtime.h>

// Hermite activation == elementwise degree-8 polynomial.
// Stage 1: fold (coefficients[9], coeffs[9x5], grid[9x5]) -> poly[9] in d_ws.
//          Branchless, lane-parallel: lane k accumulates all terms with
//          grid==k via select+FMA (no m0/movrel indexed writes, no branches).
// Stage 2: streaming Horner: b128 NT loads/stores + global_prefetch_b8.
//          Roofline: 67 MB @ 23.3 TB/s => ~2.9 us floor; 8 FMA/elem is far
//          below the VALU budget at that duration -> memory bound by design.

#define DEG   8
#define NJ    5      // DEG/2 + 1
#define NPOLY 9      // DEG + 1

typedef float v4f __attribute__((ext_vector_type(4)));

// ---------------------------------------------------------------------------
// Stage 1: one wave32. Lane k (k < 9) computes
//   poly[k] = sum over (i,j) with grid[i,j]==k of coefficients[i]*coeffs[i,j].
// Invalid (i,j) slots have coeffs==0 (and grid==0), contributing 0.
// All loads are wave-uniform -> scalarize to s_load; the accumulate is a
// fully unrolled v_cndmask + v_fma chain. No branches, no indexed VGPRs.
// ---------------------------------------------------------------------------
__global__ __launch_bounds__(32) void hermite_fold_poly_kernel(
    const float* __restrict__ coefficients,
    const float* __restrict__ coeffs,
    const float* __restrict__ gridv,
    float* __restrict__ poly) {
    const int lane = threadIdx.x;      // 0..31; lanes 0..8 are live producers
    float acc = 0.0f;
#pragma unroll
    for (int i = 0; i <= DEG; ++i) {
        const float ci = coefficients[i];
#pragma unroll
        for (int j = 0; j < NJ; ++j) {
            const float c = coeffs[i * NJ + j];
            const int   k = (int)(gridv[i * NJ + j] + 0.5f); // exact small ints
            const float t = (k == lane) ? c : 0.0f;          // v_cndmask
            acc = fmaf(ci, t, acc);                          // v_fma_f32
        }
    }
    if (lane < NPOLY) poly[lane] = acc;
}

// ---------------------------------------------------------------------------
// Stage 2: bandwidth-bound Horner evaluation.
// ---------------------------------------------------------------------------
__device__ __forceinline__ float horner9(float x, const float p[NPOLY]) {
    float y = p[DEG];
#pragma unroll
    for (int k = DEG - 1; k >= 0; --k) y = fmaf(y, x, p[k]);
    return y;
}

template <int ITERS>
__global__ __launch_bounds__(256) void hermite_horner_kernel(
    const v4f* __restrict__ x, v4f* __restrict__ out,
    const float* __restrict__ poly, int n4) {
    // 9 uniform coefficient loads, amortized over ITERS float4s per thread.
    float p[NPOLY];
#pragma unroll
    for (int k = 0; k < NPOLY; ++k) p[k] = poly[k];

    const int stride = gridDim.x * blockDim.x;
    const int base   = blockIdx.x * blockDim.x + threadIdx.x;

#pragma unroll
    for (int it = 0; it < ITERS; ++it) {
        const int i = base + it * stride;
        if (i < n4) {
            // gfx1250: global_prefetch_b8 for the next grid-stride tile.
            if (it + 1 < ITERS && (i + stride) < n4)
                __builtin_prefetch((const void*)&x[i + stride], 0, 1);

            // Streamed once, never re-read: non-temporal b128 load.
            v4f v = __builtin_nontemporal_load(&x[i]);
            v4f r;
            r.x = horner9(v.x, p);
            r.y = horner9(v.y, p);
            r.z = horner9(v.z, p);
            r.w = horner9(v.w, p);
            // Write-once output: non-temporal b128 store.
            __builtin_nontemporal_store(r, &out[i]);
        }
    }
}

// Scalar tail (n % 4 != 0). Not launched for the 2048x4096 case.
__global__ void hermite_tail_kernel(const float* __restrict__ x,
                                    float* __restrict__ out,
                                    const float* __restrict__ poly,
                                    int start, int n) {
    float p[NPOLY];
#pragma unroll
    for (int k = 0; k < NPOLY; ++k) p[k] = poly[k];
    int i = start + blockIdx.x * blockDim.x + threadIdx.x;
    if (i < n) out[i] = horner9(x[i], p);
}

// ---------------------------------------------------------------------------
extern "C" void kernel_launch(void* const* d_in, const int* in_sizes, int n_in,
                              void* d_out, int out_size, void* d_ws, size_t ws_size,
                              hipStream_t stream) {
    const float* x            = (const float*)d_in[0];
    const float* coefficients = (const float*)d_in[1];
    const float* coeffs       = (const float*)d_in[2];
    const float* gridv        = (const float*)d_in[3];
    float*       out          = (float*)d_out;
    float*       poly         = (float*)d_ws;   // 9 floats of scratch

    const int n    = in_sizes[0];               // 2048*4096 = 8388608
    const int n4   = n / 4;
    const int tail = n - n4 * 4;

    hermite_fold_poly_kernel<<<1, 32, 0, stream>>>(coefficients, coeffs, gridv, poly);

    constexpr int ITERS   = 4;
    constexpr int THREADS = 256;                // 8 wave32 waves per block
    const int blocks = (n4 + THREADS * ITERS - 1) / (THREADS * ITERS); // 2048
    if (n4 > 0) {
        hermite_horner_kernel<ITERS><<<blocks, THREADS, 0, stream>>>(
            (const v4f*)x, (v4f*)out, poly, n4);
    }
    if (tail > 0) {
        hermite_tail_kernel<<<1, 64, 0, stream>>>(x, out, poly, n4 * 4, n);
    }
}